// Kernel_56581899157823
// MI455X (gfx1250) — compile-verified
//
#include <hip/hip_runtime.h>

typedef __attribute__((ext_vector_type(2))) float v2f;
typedef __attribute__((ext_vector_type(8))) float v8f;

#define KS 8     // INSIZE
#define HS 64    // HSIZE

// Each wave computes a 16-row tile: out[16,64] = vs[16,8] @ W[8,64] + colsum(b).
// Uses V_WMMA_F32_16X16X4_F32: two K-slices (K=0..3, K=4..7) x four N-tiles.
__global__ void __launch_bounds__(256) msg_wmma_kernel(
    const float* __restrict__ vs,
    const float* __restrict__ W,
    const float* __restrict__ bias,
    float* __restrict__ out,
    int ntiles)
{
    const int lane          = threadIdx.x & 31;
    const int waveInBlock   = threadIdx.x >> 5;
    const int wavesPerBlock = blockDim.x >> 5;
    const int gwave  = blockIdx.x * wavesPerBlock + waveInBlock;
    const int nwaves = gridDim.x * wavesPerBlock;

    const int half = lane >> 4;   // 0: lanes 0-15, 1: lanes 16-31
    const int l15  = lane & 15;

    // ---- B operands (W tiles): loop-invariant, kept in VGPRs ----
    // B 4x16 f32 layout: VGPR v, lane -> K = 4*s + 2*half + v, N = l15.
    v2f Bop[2][4];
#pragma unroll
    for (int s = 0; s < 2; ++s) {
#pragma unroll
        for (int n = 0; n < 4; ++n) {
            const int krow = 4 * s + 2 * half;
            const int col  = 16 * n + l15;
            v2f b2;
            b2.x = W[krow * HS + col];
            b2.y = W[(krow + 1) * HS + col];
            Bop[s][n] = b2;
        }
    }

    // ---- bias column sums: one scalar per N-tile (same for every C VGPR) ----
    float bs[4];
#pragma unroll
    for (int n = 0; n < 4; ++n) {
        const int col = 16 * n + l15;
        float acc = 0.0f;
#pragma unroll
        for (int i = 0; i < KS; ++i) acc += bias[i * HS + col];
        bs[n] = acc;
    }

    // ---- grid-stride over 16-row tiles (wave-uniform: EXEC stays all-ones) ----
    for (int t = gwave; t < ntiles; t += nwaves) {
        const long rowbase = (long)t * 16;

        // A 16x4 f32 layout: lane -> row = l15, K = 2*half + v (+4 for slice 1)
        const float* arow = vs + (rowbase + l15) * KS + 2 * half;
        const v2f A0 = __builtin_nontemporal_load((const v2f*)(arow));      // K 0..3
        const v2f A1 = __builtin_nontemporal_load((const v2f*)(arow + 4));  // K 4..7

        float* obase = out + rowbase * HS + (long)(8 * half) * HS + l15;
#pragma unroll
        for (int n = 0; n < 4; ++n) {
            v8f c;
#pragma unroll
            for (int i = 0; i < 8; ++i) c[i] = bs[n];
            // (neg_a, A, neg_b, B, c_mod, C, reuse_a, reuse_b)
            c = __builtin_amdgcn_wmma_f32_16x16x4_f32(
                    false, A0, false, Bop[0][n], (short)0, c, false, false);
            c = __builtin_amdgcn_wmma_f32_16x16x4_f32(
                    false, A1, false, Bop[1][n], (short)0, c, false, false);

            // D layout: VGPR i -> row = i + 8*half, col = 16*n + l15
            float* op = obase + 16 * n;
#pragma unroll
            for (int i = 0; i < 8; ++i)
                __builtin_nontemporal_store(c[i], op + (long)i * HS);
        }
    }
}

// Scalar fallback for a partial tail tile (<16 rows). Unused when B % 16 == 0.
__global__ void msg_tail_kernel(const float* __restrict__ vs,
                                const float* __restrict__ W,
                                const float* __restrict__ bias,
                                float* __restrict__ out,
                                long rowstart, int nrows)
{
    const int idx = blockIdx.x * blockDim.x + threadIdx.x;
    if (idx >= nrows * HS) return;
    const long r = rowstart + idx / HS;
    const int  h = idx % HS;
    float acc = 0.0f;
#pragma unroll
    for (int i = 0; i < KS; ++i)
        acc += vs[r * KS + i] * W[i * HS + h] + bias[i * HS + h];
    out[r * HS + h] = acc;
}

extern "C" void kernel_launch(void* const* d_in, const int* in_sizes, int n_in,
                              void* d_out, int out_size, void* d_ws, size_t ws_size,
                              hipStream_t stream)
{
    const float* vs   = (const float*)d_in[0];  // [B, 8]
    const float* W    = (const float*)d_in[1];  // [8, 64]
    const float* bias = (const float*)d_in[2];  // [8, 64]
    float* out        = (float*)d_out;          // [B, 64]

    const long B      = (long)in_sizes[0] / KS;
    const int  ntiles = (int)(B / 16);
    const int  rem    = (int)(B - (long)ntiles * 16);

    if (ntiles > 0) {
        const int wavesPerBlock = 8;              // 256 threads
        int nblocks = (ntiles + wavesPerBlock - 1) / wavesPerBlock;
        if (nblocks > 2048) nblocks = 2048;       // grid-stride covers the rest
        msg_wmma_kernel<<<dim3(nblocks), dim3(256), 0, stream>>>(
            vs, W, bias, out, ntiles);
    }
    if (rem > 0) {
        const int n = rem * HS;
        msg_tail_kernel<<<dim3((n + 255) / 256), dim3(256), 0, stream>>>(
            vs, W, bias, out, (long)ntiles * 16, rem);
    }
}